// BoundaryLoss_13984413516001
// MI455X (gfx1250) — compile-verified
//
#include <hip/hip_runtime.h>
#include <hip/hip_bf16.h>
#include <cfloat>

typedef __attribute__((ext_vector_type(2))) float v2f;
typedef __attribute__((ext_vector_type(8))) float v8f;

#define NBATCH 32
#define NWAY   256
#define NBND   4096
#define NTILES   (NBND / 16)            // 256 boundary tiles of 16
#define WAVES_PER_BLK 4                 // 128 threads
#define BLKS_PER_B    4                 // 4 blocks per batch
#define NBLOCKS  (NBATCH * BLKS_PER_B)  // 128
#define WG_PER_B (NWAY / 16)            // 16 waypoint groups per batch
#define ALPHA 1.0f
#define BETA  0.5f

// Block = 4 wave32's, all working on one batch. Boundary points for the batch
// are staged once into LDS as {x, y, z, ||b||^2} (64 KB). Each wave owns one
// 16-waypoint group and sweeps all 4096 boundary points in 16-wide tiles with
// V_WMMA_F32_16X16X4_F32 computing  ||b||^2 - 2 w.b  (argmin-equivalent to
// squared distance; the row-constant ||w||^2 is dropped).
__global__ __launch_bounds__(WAVES_PER_BLK * 32) void boundary_argmin_kernel(
    const float* __restrict__ wpts,   // [B, W, 3]
    const float* __restrict__ bpts,   // [B, N, 3]
    const float* __restrict__ bnrm,   // [B, N, 3]
    float* __restrict__ partial)      // [B * WG_PER_B]
{
    __shared__ float4 spts[NBND];     // 64 KB: {x, y, z, ||b||^2}

    const int bid  = blockIdx.x;                 // 0..127
    const int b    = bid / BLKS_PER_B;
    const int sub  = bid % BLKS_PER_B;
    const int tid  = threadIdx.x;                // 0..127
    const int wv   = tid >> 5;                   // wave in block 0..3
    const int lane = tid & 31;
    const int col  = lane & 15;
    const bool hi  = lane >= 16;
    const int g    = sub * WAVES_PER_BLK + wv;   // waypoint group 0..15

    // ---- stage boundary points (coalesced: lane-stride over points) --------
    const float* bbase = bpts + (size_t)b * NBND * 3;
#pragma unroll
    for (int i = 0; i < NBND / (WAVES_PER_BLK * 32); ++i) {
        const int p = tid + i * (WAVES_PER_BLK * 32);
        const float x = bbase[p * 3 + 0];
        const float y = bbase[p * 3 + 1];
        const float z = bbase[p * 3 + 2];
        spts[p] = make_float4(x, y, z, x * x + y * y + z * z);
    }
    __syncthreads();

    // ---- A matrix: 16x4, row m = waypoint g*16+m, vector [-2x,-2y,-2z,1]
    // 32-bit A 16x4 layout: lanes 0-15 hold K=0,1 ; lanes 16-31 hold K=2,3.
    const float* wp = wpts + ((size_t)b * NWAY + g * 16 + col) * 3;
    v2f a;
    a.x = hi ? (-2.0f * wp[2]) : (-2.0f * wp[0]);
    a.y = hi ? 1.0f            : (-2.0f * wp[1]);

    float minv[8];
    int   mini[8];
#pragma unroll
    for (int r = 0; r < 8; ++r) { minv[r] = FLT_MAX; mini[r] = 0; }

    // ---- hot loop: ds_load_b128 + branchless B build + WMMA + argmin ------
#pragma unroll 4
    for (int t = 0; t < NTILES; ++t) {
        const int n = t * 16 + col;
        const float4 p = spts[n];
        // B matrix: 4x16, col n, vector [bx,by,bz,||b||^2]
        // lanes 0-15 hold K=0,1 ; lanes 16-31 hold K=2,3 (branchless selects)
        v2f bv;
        bv.x = hi ? p.z : p.x;
        bv.y = hi ? p.w : p.y;

        v8f c = {};
        c = __builtin_amdgcn_wmma_f32_16x16x4_f32(
                /*neg_a=*/false, a, /*neg_b=*/false, bv,
                /*c_mod=*/(short)0, c, /*reuse_a=*/false, /*reuse_b=*/false);

        // D layout: lane = column, VGPR r = row r (lanes 0-15) / 8+r (16-31)
#pragma unroll
        for (int r = 0; r < 8; ++r) {
            const float v = c[r];
            if (v < minv[r]) { minv[r] = v; mini[r] = n; }
        }
    }

    // ---- argmin reduction across the 16 columns (within each half-wave) ---
#pragma unroll
    for (int m = 1; m <= 8; m <<= 1) {
#pragma unroll
        for (int r = 0; r < 8; ++r) {
            const float ov = __shfl_xor(minv[r], m, 32);
            const int   oi = __shfl_xor(mini[r], m, 32);
            if (ov < minv[r] || (ov == minv[r] && oi < mini[r])) {
                minv[r] = ov; mini[r] = oi;
            }
        }
    }

    // ---- lanes 0 (rows 0-7) and 16 (rows 8-15) finish their 8 waypoints ---
    float psum = 0.0f;
    if (lane == 0 || lane == 16) {
        const float* nbase = bnrm + (size_t)b * NBND * 3;
        const int rowbase = hi ? 8 : 0;
#pragma unroll
        for (int r = 0; r < 8; ++r) {
            const int row = rowbase + r;
            const float* w2 = wpts + ((size_t)b * NWAY + g * 16 + row) * 3;
            const int idx = mini[r];
            const float4 cp = spts[idx];          // closest point from LDS
            const float* cn = nbase + (size_t)idx * 3;
            const float dx = w2[0] - cp.x;
            const float dy = w2[1] - cp.y;
            const float dz = w2[2] - cp.z;
            const float dp = dx * cn[0] + dy * cn[1] + dz * cn[2];
            psum += (dp > 0.0f) ? (ALPHA * dp) : (__expf(BETA * dp) - 1.0f);
        }
    }
    const float s16 = __shfl(psum, 16, 32);
    if (lane == 0) partial[b * WG_PER_B + g] = psum + s16;
}

__global__ __launch_bounds__(512) void boundary_reduce_kernel(
    const float* __restrict__ partial, float* __restrict__ out)
{
    __shared__ float s[NBATCH * WG_PER_B];
    const int t = threadIdx.x;
    s[t] = partial[t];
    __syncthreads();
    for (int off = (NBATCH * WG_PER_B) / 2; off > 0; off >>= 1) {
        if (t < off) s[t] += s[t + off];
        __syncthreads();
    }
    if (t == 0) out[0] = s[0] * (1.0f / (float)(NBATCH * NWAY));
}

extern "C" void kernel_launch(void* const* d_in, const int* in_sizes, int n_in,
                              void* d_out, int out_size, void* d_ws, size_t ws_size,
                              hipStream_t stream) {
    const float* waypoints      = (const float*)d_in[0];  // [32,256,3]
    const float* boundarypoints = (const float*)d_in[1];  // [32,4096,3]
    const float* boundarynormal = (const float*)d_in[2];  // [32,4096,3]
    float* out     = (float*)d_out;
    float* partial = (float*)d_ws;                        // 512 floats

    boundary_argmin_kernel<<<NBLOCKS, WAVES_PER_BLK * 32, 0, stream>>>(
        waypoints, boundarypoints, boundarynormal, partial);
    boundary_reduce_kernel<<<1, NBATCH * WG_PER_B, 0, stream>>>(partial, out);
}